// IMUPreintegrator_47528108097640
// MI455X (gfx1250) — compile-verified
//
#include <hip/hip_runtime.h>
#include <math.h>

// One wave32 per batch. Covariance kept as a zero-padded 16x16 FP32 tile.
// Per step: 8x V_WMMA_F32_16X16X4_F32 on the matrix pipe, LDS tile round-trip
// for the C/D-layout <-> A-layout transpose, scalar SO(3) math replicated
// across lanes (uniform -> SMEM broadcast).

#define WAVES 4   // waves (=batches) per block; 2048/4 = 512 blocks

typedef __attribute__((ext_vector_type(2))) float v2f;
typedef __attribute__((ext_vector_type(8))) float v8f;

#define DEV static __device__ __forceinline__

struct Q4 { float x, y, z, w; };

DEV Q4 qmul(Q4 a, Q4 b) {
  Q4 r;
  r.x = a.w * b.x + a.x * b.w + a.y * b.z - a.z * b.y;
  r.y = a.w * b.y - a.x * b.z + a.y * b.w + a.z * b.x;
  r.z = a.w * b.z + a.x * b.y - a.y * b.x + a.z * b.w;
  r.w = a.w * b.w - a.x * b.x - a.y * b.y - a.z * b.z;
  return r;
}

DEV void q2m(Q4 q, float* R) {
  float x = q.x, y = q.y, z = q.z, w = q.w;
  R[0] = 1.f - 2.f * (y * y + z * z); R[1] = 2.f * (x * y - z * w); R[2] = 2.f * (x * z + y * w);
  R[3] = 2.f * (x * y + z * w); R[4] = 1.f - 2.f * (x * x + z * z); R[5] = 2.f * (y * z - x * w);
  R[6] = 2.f * (x * z - y * w); R[7] = 2.f * (y * z + x * w); R[8] = 1.f - 2.f * (x * x + y * y);
}

// v' = R(q) v  (unit quaternion rotate)
DEV void qrot(Q4 q, float vx, float vy, float vz, float& ox, float& oy, float& oz) {
  float tx = q.y * vz - q.z * vy + q.w * vx;
  float ty = q.z * vx - q.x * vz + q.w * vy;
  float tz = q.x * vy - q.y * vx + q.w * vz;
  ox = vx + 2.f * (q.y * tz - q.z * ty);
  oy = vy + 2.f * (q.z * tx - q.x * tz);
  oz = vz + 2.f * (q.x * ty - q.y * tx);
}

// D = A(16x4 f32) x B(4x16 f32) + C(16x16 f32)
DEV v8f wmma4(v2f a, v2f b, v8f c) {
  return __builtin_amdgcn_wmma_f32_16x16x4_f32(
      /*neg_a=*/false, a, /*neg_b=*/false, b,
      /*c_mod=*/(short)0, c, /*reuse_a=*/false, /*reuse_b=*/false);
}

// State-transition matrix element A[r][c] (9x9, zero-padded to 16x16).
// blk layout (LDS): [0..8]=drT, [9..17]=RHa, [18..26]=Jr*dt, [27..35]=Rnew
DEV float Aelem(const float* blk, float dtk, int r, int c) {
  float v = 0.f;
  if (r < 3) {
    if (c < 3) v = blk[r * 3 + c];                                   // drT
  } else if (r < 6) {
    if (c < 3)      v = -blk[9 + (r - 3) * 3 + c] * dtk;             // -RHa*dt
    else if (c < 6) v = (c - 3 == r - 3) ? 1.f : 0.f;                // I
  } else if (r < 9) {
    if (c < 3)      v = -0.5f * dtk * dtk * blk[9 + (r - 6) * 3 + c];// -.5*RHa*dt^2
    else if (c < 6) v = (c - 3 == r - 6) ? dtk : 0.f;                // I*dt
    else if (c < 9) v = (c - 6 == r - 6) ? 1.f : 0.f;                // I
  }
  return v;
}

DEV float Bgelem(const float* blk, int r, int c) {                   // [JrDt;0;0]
  return (r < 3 && c < 3) ? blk[18 + r * 3 + c] : 0.f;
}

DEV float Baelem(const float* blk, float dtk, int r, int c) {        // [0;R*dt;.5R*dt^2]
  if (c >= 3) return 0.f;
  if (r >= 3 && r < 6) return blk[27 + (r - 3) * 3 + c] * dtk;
  if (r >= 6 && r < 9) return 0.5f * dtk * dtk * blk[27 + (r - 6) * 3 + c];
  return 0.f;
}

__global__ __launch_bounds__(WAVES * 32) void imu_preint_kernel(
    const float* __restrict__ dtp,  const float* __restrict__ angp,
    const float* __restrict__ accp, const float* __restrict__ pos0p,
    const float* __restrict__ vel0p, const float* __restrict__ rot0p,
    float* __restrict__ out, int Bn, int Nn)
{
  // per-wave: cov tile (256) + T tile (256) + block scratch (64)
  __shared__ float smem[WAVES][576];
  const int lane = (int)(threadIdx.x & 31u);
  const int wv   = (int)(threadIdx.x >> 5);
  const int b    = (int)blockIdx.x * WAVES + wv;
  if (b >= Bn) return;   // wave-uniform

  float* covT = smem[wv];
  float* tT   = smem[wv] + 256;
  float* blk  = smem[wv] + 512;

  const int n16 = lane & 15;       // N (B/C/D striping) and M (A striping)
  const int khi = lane >> 4;       // 0: K=0,1 half; 1: K=2,3 half

  // cov starts at zero; padding rows/cols must be zero too
#pragma unroll
  for (int j = 0; j < 8; ++j) covT[lane + 32 * j] = 0.0f;
  asm volatile("s_wait_dscnt 0" ::: "memory");

  const float G   = 9.81007f;
  const float SG2 = 1.6968e-4f * 1.6968e-4f;
  const float SA2 = 2.0e-3f * 2.0e-3f;

  float dpx = 0.f, dpy = 0.f, dpz = 0.f;
  float dvx = 0.f, dvy = 0.f, dvz = 0.f;
  float ta  = 0.f;
  Q4 drq = {0.f, 0.f, 0.f, 1.f};
  Q4 r0  = { rot0p[4 * b + 0], rot0p[4 * b + 1], rot0p[4 * b + 2], rot0p[4 * b + 3] };

  const float* dtB  = dtp  + (size_t)b * Nn;
  const float* angB = angp + (size_t)b * Nn * 3;
  const float* accB = accp + (size_t)b * Nn * 3;

  for (int k = 0; k < Nn; ++k) {
    if (k + 48 < Nn) {   // stream ahead: global_prefetch_b8
      __builtin_prefetch(accB + 3 * (k + 48), 0, 1);
      __builtin_prefetch(angB + 3 * (k + 48), 0, 1);
      __builtin_prefetch(dtB + (k + 48), 0, 1);
    }
    float dtk = dtB[k];
    float gx = angB[3 * k + 0], gy = angB[3 * k + 1], gz = angB[3 * k + 2];
    float ax = accB[3 * k + 0], ay = accB[3 * k + 1], az = accB[3 * k + 2];

    // --- SO(3) exp of phi = ang*dt ---
    float phx = gx * dtk, phy = gy * dtk, phz = gz * dtk;
    float t2 = phx * phx + phy * phy + phz * phz;
    bool  sm = t2 < 1e-8f;
    float t2s = sm ? 1.f : t2;
    float t  = sqrtf(t2s);
    float sh = sm ? (0.5f - t2 * (1.f / 48.f)) : (sinf(0.5f * t) / t);
    float wq = sm ? (1.f - t2 * 0.125f) : cosf(0.5f * t);
    Q4 dr = { phx * sh, phy * sh, phz * sh, wq };

    // a = acc - R(rot0*drq*dr)^T g   (only row 2 of R needed)
    Q4 qtot = qmul(qmul(r0, drq), dr);
    float r20 = 2.f * (qtot.x * qtot.z - qtot.y * qtot.w);
    float r21 = 2.f * (qtot.y * qtot.z + qtot.x * qtot.w);
    float r22 = 1.f - 2.f * (qtot.x * qtot.x + qtot.y * qtot.y);
    float avx = ax - r20 * G, avy = ay - r21 * G, avz = az - r22 * G;

    // Ra = R(drq_old) a ; integrate dp, dv
    float Rax, Ray, Raz;
    qrot(drq, avx, avy, avz, Rax, Ray, Raz);
    float hdt2 = 0.5f * dtk * dtk;
    dpx += dvx * dtk + Rax * hdt2;  dpy += dvy * dtk + Ray * hdt2;  dpz += dvz * dtk + Raz * hdt2;
    dvx += Rax * dtk;               dvy += Ray * dtk;               dvz += Raz * dtk;
    Q4 drq_n = qmul(drq, dr);
    ta += dtk;

    // --- covariance blocks ---
    float Rd[9]; q2m(dr, Rd);       // drT = Rd^T at store time
    float Rn[9]; q2m(drq_n, Rn);
    float RHa[9];                    // Rn @ skew(acc)
#pragma unroll
    for (int i = 0; i < 3; ++i) {
      RHa[i * 3 + 0] = Rn[i * 3 + 1] * az - Rn[i * 3 + 2] * ay;
      RHa[i * 3 + 1] = Rn[i * 3 + 2] * ax - Rn[i * 3 + 0] * az;
      RHa[i * 3 + 2] = Rn[i * 3 + 0] * ay - Rn[i * 3 + 1] * ax;
    }
    // right Jacobian: Jr = (1-c2*t2) I - c1 K + c2 p p^T , scaled by dt
    float c1 = sm ? (0.5f - t2 * (1.f / 24.f)) : ((1.f - cosf(t)) / t2s);
    float c2 = sm ? (1.f / 6.f - t2 * (1.f / 120.f)) : ((t - sinf(t)) / (t2s * t));
    float om = 1.f - c2 * t2;
    float Jd[9];
    Jd[0] = (om + c2 * phx * phx) * dtk;
    Jd[1] = ( c1 * phz + c2 * phx * phy) * dtk;
    Jd[2] = (-c1 * phy + c2 * phx * phz) * dtk;
    Jd[3] = (-c1 * phz + c2 * phy * phx) * dtk;
    Jd[4] = (om + c2 * phy * phy) * dtk;
    Jd[5] = ( c1 * phx + c2 * phy * phz) * dtk;
    Jd[6] = ( c1 * phy + c2 * phz * phx) * dtk;
    Jd[7] = (-c1 * phx + c2 * phz * phy) * dtk;
    Jd[8] = (om + c2 * phz * phz) * dtk;

    if (lane == 0) {
#pragma unroll
      for (int i = 0; i < 3; ++i)
#pragma unroll
        for (int j = 0; j < 3; ++j) blk[i * 3 + j] = Rd[j * 3 + i];   // drT
#pragma unroll
      for (int i = 0; i < 9; ++i) blk[9 + i]  = RHa[i];
#pragma unroll
      for (int i = 0; i < 9; ++i) blk[18 + i] = Jd[i];
#pragma unroll
      for (int i = 0; i < 9; ++i) blk[27 + i] = Rn[i];
    }
    asm volatile("s_wait_dscnt 0" ::: "memory");

    // ---- product 1: T = A @ cov  (3 WMMAs over K=0..11) ----
    v8f tacc;
#pragma unroll
    for (int j = 0; j < 8; ++j) tacc[j] = 0.f;
#pragma unroll
    for (int kb = 0; kb < 12; kb += 4) {
      int kk = kb + 2 * khi;
      v2f a, bb;
      a.x  = Aelem(blk, dtk, n16, kk);
      a.y  = Aelem(blk, dtk, n16, kk + 1);
      bb.x = covT[kk * 16 + n16];
      bb.y = covT[(kk + 1) * 16 + n16];
      tacc = wmma4(a, bb, tacc);
    }
#pragma unroll
    for (int j = 0; j < 8; ++j) tT[(j + 8 * khi) * 16 + n16] = tacc[j];
    asm volatile("s_wait_dscnt 0" ::: "memory");

    // ---- product 2: cov' = T @ A^T + SG2*Bg@Bg^T + SA2*Ba@Ba^T ----
    v8f dacc;
#pragma unroll
    for (int j = 0; j < 8; ++j) dacc[j] = 0.f;
#pragma unroll
    for (int kb = 0; kb < 12; kb += 4) {
      int kk = kb + 2 * khi;
      v2f a, bb;
      a.x  = tT[n16 * 16 + kk];
      a.y  = tT[n16 * 16 + kk + 1];
      bb.x = Aelem(blk, dtk, n16, kk);       // A^T[k][n] = A[n][k]
      bb.y = Aelem(blk, dtk, n16, kk + 1);
      dacc = wmma4(a, bb, dacc);
    }
    {
      int kk = 2 * khi;
      v2f a, bb;
      bb.x = Bgelem(blk, n16, kk);
      bb.y = Bgelem(blk, n16, kk + 1);
      a.x  = SG2 * bb.x;                     // fold sigma into A operand
      a.y  = SG2 * bb.y;
      dacc = wmma4(a, bb, dacc);
      bb.x = Baelem(blk, dtk, n16, kk);
      bb.y = Baelem(blk, dtk, n16, kk + 1);
      a.x  = SA2 * bb.x;
      a.y  = SA2 * bb.y;
      dacc = wmma4(a, bb, dacc);
    }
#pragma unroll
    for (int j = 0; j < 8; ++j) covT[(j + 8 * khi) * 16 + n16] = dacc[j];
    asm volatile("s_wait_dscnt 0" ::: "memory");

    drq = drq_n;
  }

  // ---- outputs: rot[B,4] | vel[B,3] | pos[B,3] | cov[B,9,9] ----
  if (lane == 0) {
    Q4 ro = qmul(r0, drq);
    out[(size_t)b * 4 + 0] = ro.x;
    out[(size_t)b * 4 + 1] = ro.y;
    out[(size_t)b * 4 + 2] = ro.z;
    out[(size_t)b * 4 + 3] = ro.w;

    float v0x = vel0p[3 * b + 0], v0y = vel0p[3 * b + 1], v0z = vel0p[3 * b + 2];
    float p0x = pos0p[3 * b + 0], p0y = pos0p[3 * b + 1], p0z = pos0p[3 * b + 2];
    float rvx, rvy, rvz, rpx, rpy, rpz;
    qrot(r0, dvx, dvy, dvz, rvx, rvy, rvz);
    qrot(r0, dpx, dpy, dpz, rpx, rpy, rpz);

    float* vout = out + (size_t)4 * Bn;
    vout[(size_t)b * 3 + 0] = v0x + rvx;
    vout[(size_t)b * 3 + 1] = v0y + rvy;
    vout[(size_t)b * 3 + 2] = v0z + rvz;

    float* pout = out + (size_t)7 * Bn;
    pout[(size_t)b * 3 + 0] = p0x + v0x * ta + rpx;
    pout[(size_t)b * 3 + 1] = p0y + v0y * ta + rpy;
    pout[(size_t)b * 3 + 2] = p0z + v0z * ta + rpz;
  }
  float* cout = out + (size_t)10 * Bn + (size_t)b * 81;
  for (int idx = lane; idx < 81; idx += 32)
    cout[idx] = covT[(idx / 9) * 16 + (idx % 9)];
}

extern "C" void kernel_launch(void* const* d_in, const int* in_sizes, int n_in,
                              void* d_out, int out_size, void* d_ws, size_t ws_size,
                              hipStream_t stream) {
  (void)n_in; (void)out_size; (void)d_ws; (void)ws_size;
  const float* dtp   = (const float*)d_in[0];
  const float* angp  = (const float*)d_in[1];
  const float* accp  = (const float*)d_in[2];
  const float* pos0p = (const float*)d_in[3];
  const float* vel0p = (const float*)d_in[4];
  const float* rot0p = (const float*)d_in[5];

  int Bn = in_sizes[3] / 3;              // pos0 is [B,3]
  int Nn = in_sizes[0] / Bn;             // dt is [B,N,1]
  int blocks = (Bn + WAVES - 1) / WAVES;
  imu_preint_kernel<<<blocks, WAVES * 32, 0, stream>>>(
      dtp, angp, accp, pos0p, vel0p, rot0p, (float*)d_out, Bn, Nn);
}